// SymmetryLoss_10050223473280
// MI455X (gfx1250) — compile-verified
//
#include <hip/hip_runtime.h>

typedef float v2f __attribute__((ext_vector_type(2)));
typedef float v8f __attribute__((ext_vector_type(8)));

#define NBLK 1024
#define TPB  256
#define WR_CONST 25.0f

// Element [r][c] (r=0..2, c=0..3, c==3 is translation) of transform t's affine matrix.
// t=0..2: plane reflections (n,d) -> (I - 2 n n^T | -2 d n)
// t=3..5: quaternion maps q p q*:  R = (w^2-|v|^2) I + 2 v v^T + 2 w [v]x, translation 0
__device__ __forceinline__ float transform_elem(const float* __restrict__ planes,
                                                const float* __restrict__ axes,
                                                int t, int r, int c) {
  if (t < 3) {
    const float nr = planes[4 * t + r];
    if (c == 3) return -2.0f * planes[4 * t + 3] * nr;
    const float nc = planes[4 * t + c];
    return ((r == c) ? 1.0f : 0.0f) - 2.0f * nr * nc;
  }
  const float* q = axes + 4 * (t - 3);
  if (c == 3) return 0.0f;
  const float w = q[0], x = q[1], y = q[2], z = q[3];
  const float v[3] = {x, y, z};
  float cr;
  if (r == c)      cr = 0.0f;
  else if (r == 0) cr = (c == 1) ? -z : y;
  else if (r == 1) cr = (c == 0) ?  z : -x;
  else             cr = (c == 0) ? -y : x;
  const float s = (r == c) ? (w * w - (x * x + y * y + z * z)) : 0.0f;
  return s + 2.0f * v[r] * v[c] + 2.0f * w * cr;
}

// Nearest-grid-cell distance. rintf == round-half-even == jnp.round.
__device__ __forceinline__ float cp_dist(float tx, float ty, float tz,
                                         const float* __restrict__ grid,
                                         float g0, float g1, float g2,
                                         float s0, float s1, float s2) {
  float fx = fminf(fmaxf(rintf((tx - g0) * s0), 0.0f), 255.0f);
  float fy = fminf(fmaxf(rintf((ty - g1) * s1), 0.0f), 255.0f);
  float fz = fminf(fmaxf(rintf((tz - g2) * s2), 0.0f), 255.0f);
  unsigned ix = (unsigned)(int)fx, iy = (unsigned)(int)fy, iz = (unsigned)(int)fz;
  const float* cp = grid + 3u * ((ix * 256u + iy) * 256u + iz);
  float dx = tx - cp[0], dy = ty - cp[1], dz = tz - cp[2];
  return sqrtf(dx * dx + dy * dy + dz * dz);
}

__global__ void __launch_bounds__(TPB) symm_main_kernel(
    const float* __restrict__ planes, const float* __restrict__ axes,
    const float* __restrict__ pts,    const float* __restrict__ grid,
    const float* __restrict__ gmin,   const float* __restrict__ gmax,
    float* __restrict__ ws, int N) {
  const int tid  = threadIdx.x;
  const int lane = tid & 31;
  const int row  = lane & 15;   // M row in A / point column in B
  const int kh   = lane >> 4;   // K-half: 0 -> K=0,1 ; 1 -> K=2,3
  const int c0   = 2 * kh, c1 = c0 + 1;

  const float g0 = gmin[0], g1 = gmin[1], g2 = gmin[2];
  const float s0 = 255.0f / (gmax[0] - g0);
  const float s1 = 255.0f / (gmax[1] - g1);
  const float s2 = 255.0f / (gmax[2] - g2);

  // A1: transforms 0..3 in row groups of 4 (rows 3,7,11,15 zero).
  // A2: transform 4 in rows 0-2, transform 5 in rows 8-10.
  v2f A1; A1.x = 0.0f; A1.y = 0.0f;
  v2f A2; A2.x = 0.0f; A2.y = 0.0f;
  {
    int grp = row >> 2, lr = row & 3;
    if (lr < 3) {
      A1.x = transform_elem(planes, axes, grp, lr, c0);
      A1.y = transform_elem(planes, axes, grp, lr, c1);
    }
  }
  if (row < 3) {
    A2.x = transform_elem(planes, axes, 4, row, c0);
    A2.y = transform_elem(planes, axes, 4, row, c1);
  } else if (row >= 8 && row < 11) {
    A2.x = transform_elem(planes, axes, 5, row - 8, c0);
    A2.y = transform_elem(planes, axes, 5, row - 8, c1);
  }

  float acc = 0.0f;
  const int fullTiles = N >> 4;          // 16-point wave tiles
  const int numPairs  = fullTiles >> 1;  // process 2 tiles (32 points) per iteration
  const int gw = blockIdx.x * (TPB / 32) + (tid >> 5);
  const int nw = gridDim.x * (TPB / 32);

  // ---- software-pipelined pair loop: prefetch next pair's points ----
  float pax = 0.f, pay = 0.f, paz = 0.f, pbx = 0.f, pby = 0.f, pbz = 0.f;
  int tp = gw;
  if (tp < numPairs) {
    const float* pp = pts + 3 * (32 * tp + row);
    pax = pp[0];  pay = pp[1];  paz = pp[2];
    pbx = pp[48]; pby = pp[49]; pbz = pp[50];
  }
  while (tp < numPairs) {
    const int nxt = tp + nw;
    float nax = 0.f, nay = 0.f, naz = 0.f, nbx = 0.f, nby = 0.f, nbz = 0.f;
    if (nxt < numPairs) {  // wave-uniform; loads overlap this iteration's gathers
      const float* pp = pts + 3 * (32 * nxt + row);
      nax = pp[0];  nay = pp[1];  naz = pp[2];
      nbx = pp[48]; nby = pp[49]; nbz = pp[50];
    }

    // B (4x16): lanes 0-15 -> (x,y), lanes 16-31 -> (z,1)
    v2f Ba; Ba.x = kh ? paz : pax; Ba.y = kh ? 1.0f : pay;
    v2f Bb; Bb.x = kh ? pbz : pbx; Bb.y = kh ? 1.0f : pby;

    v8f cz = {};
    v8f d1a = __builtin_amdgcn_wmma_f32_16x16x4_f32(false, A1, false, Ba, (short)0, cz, false, false);
    v8f d2a = __builtin_amdgcn_wmma_f32_16x16x4_f32(false, A2, false, Ba, (short)0, cz, false, false);
    v8f d1b = __builtin_amdgcn_wmma_f32_16x16x4_f32(false, A1, false, Bb, (short)0, cz, false, false);
    v8f d2b = __builtin_amdgcn_wmma_f32_16x16x4_f32(false, A2, false, Bb, (short)0, cz, false, false);

    // Each lane owns 3 complete transformed points per tile:
    // half0: t0(d1[0..2]), t1(d1[4..6]), t4(d2[0..2]); half1: t2, t3, t5.
    // 6 independent gathers per lane per iteration -> high MLP.
    acc += cp_dist(d1a[0], d1a[1], d1a[2], grid, g0, g1, g2, s0, s1, s2);
    acc += cp_dist(d1a[4], d1a[5], d1a[6], grid, g0, g1, g2, s0, s1, s2);
    acc += cp_dist(d2a[0], d2a[1], d2a[2], grid, g0, g1, g2, s0, s1, s2);
    acc += cp_dist(d1b[0], d1b[1], d1b[2], grid, g0, g1, g2, s0, s1, s2);
    acc += cp_dist(d1b[4], d1b[5], d1b[6], grid, g0, g1, g2, s0, s1, s2);
    acc += cp_dist(d2b[0], d2b[1], d2b[2], grid, g0, g1, g2, s0, s1, s2);

    pax = nax; pay = nay; paz = naz;
    pbx = nbx; pby = nby; pbz = nbz;
    tp = nxt;
  }

  // ---- odd leftover full tile (fullTiles odd): wave 0 of block 0, WMMA path ----
  if ((fullTiles & 1) && blockIdx.x == 0 && (tid >> 5) == 0) {
    const int tile = fullTiles - 1;
    const float* pp = pts + 3 * (tile * 16 + row);
    float px = pp[0], py = pp[1], pz = pp[2];
    v2f B; B.x = kh ? pz : px; B.y = kh ? 1.0f : py;
    v8f cz = {};
    v8f d1 = __builtin_amdgcn_wmma_f32_16x16x4_f32(false, A1, false, B, (short)0, cz, false, false);
    v8f d2 = __builtin_amdgcn_wmma_f32_16x16x4_f32(false, A2, false, B, (short)0, cz, false, false);
    acc += cp_dist(d1[0], d1[1], d1[2], grid, g0, g1, g2, s0, s1, s2);
    acc += cp_dist(d1[4], d1[5], d1[6], grid, g0, g1, g2, s0, s1, s2);
    acc += cp_dist(d2[0], d2[1], d2[2], grid, g0, g1, g2, s0, s1, s2);
  }

  // ---- scalar tail (N % 16 points), block 0 only. Dead for N = 1e6. ----
  if (blockIdx.x == 0) {
    for (int i = fullTiles * 16 + tid; i < N; i += TPB) {
      float px = pts[3 * i], py = pts[3 * i + 1], pz = pts[3 * i + 2];
      for (int t = 0; t < 6; ++t) {
        float tc[3];
        for (int r = 0; r < 3; ++r) {
          tc[r] = transform_elem(planes, axes, t, r, 0) * px +
                  transform_elem(planes, axes, t, r, 1) * py +
                  transform_elem(planes, axes, t, r, 2) * pz +
                  transform_elem(planes, axes, t, r, 3);
        }
        acc += cp_dist(tc[0], tc[1], tc[2], grid, g0, g1, g2, s0, s1, s2);
      }
    }
  }

  __shared__ float red[TPB];
  red[tid] = acc;
  __syncthreads();
  for (int off = TPB / 2; off > 0; off >>= 1) {
    if (tid < off) red[tid] += red[tid + off];
    __syncthreads();
  }
  if (tid == 0) ws[blockIdx.x] = red[0];
}

__global__ void __launch_bounds__(256) symm_finalize_kernel(
    const float* __restrict__ ws, int nblk,
    const float* __restrict__ planes, const float* __restrict__ axes,
    float* __restrict__ out, float invN) {
  __shared__ float s[256];
  float a = 0.0f;
  for (int i = threadIdx.x; i < nblk; i += 256) a += ws[i];
  s[threadIdx.x] = a;
  __syncthreads();
  for (int off = 128; off > 0; off >>= 1) {
    if (threadIdx.x < off) s[threadIdx.x] += s[threadIdx.x + off];
    __syncthreads();
  }
  if (threadIdx.x == 0) {
    float loss_sd = s[0] * invN;
    // A = n n^T - I  (unnormalized plane normals)
    float sA = 0.0f;
    for (int i = 0; i < 3; ++i)
      for (int j = 0; j < 3; ++j) {
        float d = planes[4 * i + 0] * planes[4 * j + 0] +
                  planes[4 * i + 1] * planes[4 * j + 1] +
                  planes[4 * i + 2] * planes[4 * j + 2] -
                  ((i == j) ? 1.0f : 0.0f);
        sA += d * d;
      }
    // B = v v^T - I with v = normalize(axes[:,1:]) (norm clamped at 1e-12)
    float vn[3][3];
    for (int i = 0; i < 3; ++i) {
      float x = axes[4 * i + 1], y = axes[4 * i + 2], z = axes[4 * i + 3];
      float nrm = fmaxf(sqrtf(x * x + y * y + z * z), 1e-12f);
      vn[i][0] = x / nrm; vn[i][1] = y / nrm; vn[i][2] = z / nrm;
    }
    float sB = 0.0f;
    for (int i = 0; i < 3; ++i)
      for (int j = 0; j < 3; ++j) {
        float d = vn[i][0] * vn[j][0] + vn[i][1] * vn[j][1] + vn[i][2] * vn[j][2] -
                  ((i == j) ? 1.0f : 0.0f);
        sB += d * d;
      }
    float loss_r = sA + sB;
    out[0] = loss_sd + WR_CONST * loss_r;  // final_loss
    out[1] = loss_sd;                      // total_loss_sd
    out[2] = loss_r;                       // total_loss_r
  }
}

extern "C" void kernel_launch(void* const* d_in, const int* in_sizes, int n_in,
                              void* d_out, int out_size, void* d_ws, size_t ws_size,
                              hipStream_t stream) {
  const float* planes = (const float*)d_in[0];  // (1,3,4)
  const float* axes   = (const float*)d_in[1];  // (1,3,4)
  const float* pts    = (const float*)d_in[2];  // (N,3)
  const float* grid   = (const float*)d_in[3];  // (256,256,256,3)
  const float* gmin   = (const float*)d_in[4];  // (3,)
  const float* gmax   = (const float*)d_in[5];  // (3,)
  const int N = in_sizes[2] / 3;
  float* ws = (float*)d_ws;

  symm_main_kernel<<<NBLK, TPB, 0, stream>>>(planes, axes, pts, grid, gmin, gmax, ws, N);
  symm_finalize_kernel<<<1, 256, 0, stream>>>(ws, NBLK, planes, axes, (float*)d_out,
                                              1.0f / (float)N);
}